// AlignmentLayer_24781961298242
// MI455X (gfx1250) — compile-verified
//
#include <hip/hip_runtime.h>

// AlignmentLayer (Kabsch) for MI455X / gfx1250.
// L=65536 structures, 256 points each, 64 alignment points.
// Memory-bound (~0.55 GB traffic -> ~24us @ 23.3 TB/s); f32 WMMA 16x16x4
// is the (x-c)@R apply engine with translation folded into a 4x3 transform.

#define L_TOT   65536
#define N_INP   256
#define N_ALIGN 64

typedef float v2f __attribute__((ext_vector_type(2)));
typedef float v8f __attribute__((ext_vector_type(8)));
typedef float v3f __attribute__((ext_vector_type(3), aligned(4)));

// ---------------------------------------------------------------------------
// Kernel 0: center the reference (64x3).  One wave.
// ---------------------------------------------------------------------------
__global__ __launch_bounds__(32) void k_refc(const float* __restrict__ ref_x,
                                             float* __restrict__ refc) {
    const int t = threadIdx.x;                       // 0..31, handles rows t, t+32
    float x0 = ref_x[t * 3 + 0], y0 = ref_x[t * 3 + 1], z0 = ref_x[t * 3 + 2];
    float x1 = ref_x[(t + 32) * 3 + 0], y1 = ref_x[(t + 32) * 3 + 1], z1 = ref_x[(t + 32) * 3 + 2];
    float sx = x0 + x1, sy = y0 + y1, sz = z0 + z1;
#pragma unroll
    for (int off = 16; off; off >>= 1) {
        sx += __shfl_xor(sx, off, 32);
        sy += __shfl_xor(sy, off, 32);
        sz += __shfl_xor(sz, off, 32);
    }
    const float mx = sx * (1.0f / 64.0f), my = sy * (1.0f / 64.0f), mz = sz * (1.0f / 64.0f);
    refc[t * 3 + 0] = x0 - mx;  refc[t * 3 + 1] = y0 - my;  refc[t * 3 + 2] = z0 - mz;
    refc[(t + 32) * 3 + 0] = x1 - mx;  refc[(t + 32) * 3 + 1] = y1 - my;  refc[(t + 32) * 3 + 2] = z1 - mz;
}

// ---------------------------------------------------------------------------
// Kernel A: per-l mean of selected points + 3x3 covariance M = sel^T * refc.
// (refc is centered, so Sum_n refc[n]=0 and no mean subtraction is needed
// inside the product.)  One wave per l, 2 gathered points per lane,
// butterfly reduction, 64B-stride b128 result record.
// ---------------------------------------------------------------------------
__global__ __launch_bounds__(256) void k_cov(const float* __restrict__ x,
                                             const float* __restrict__ refc,
                                             const int* __restrict__ aidx,
                                             float* __restrict__ Mc) {
    const int lane = threadIdx.x & 31;
    const int l    = blockIdx.x * 8 + (threadIdx.x >> 5);
    const float* xl = x + (size_t)l * (N_INP * 3);

    float sx = 0.f, sy = 0.f, sz = 0.f;
    float m00 = 0.f, m01 = 0.f, m02 = 0.f;
    float m10 = 0.f, m11 = 0.f, m12 = 0.f;
    float m20 = 0.f, m21 = 0.f, m22 = 0.f;
#pragma unroll
    for (int j = 0; j < 2; ++j) {
        const int n  = lane + j * 32;
        const int id = aidx[n];
        const float* p = xl + id * 3;
        const float px = p[0], py = p[1], pz = p[2];
        const float* r = refc + n * 3;
        const float rx = r[0], ry = r[1], rz = r[2];
        sx += px; sy += py; sz += pz;
        m00 += px * rx; m01 += px * ry; m02 += px * rz;
        m10 += py * rx; m11 += py * ry; m12 += py * rz;
        m20 += pz * rx; m21 += pz * ry; m22 += pz * rz;
    }
#define RED_(v) v += __shfl_xor(v, off, 32);
#pragma unroll
    for (int off = 16; off; off >>= 1) {
        RED_(sx) RED_(sy) RED_(sz)
        RED_(m00) RED_(m01) RED_(m02)
        RED_(m10) RED_(m11) RED_(m12)
        RED_(m20) RED_(m21) RED_(m22)
    }
#undef RED_
    if (lane == 0) {
        float4* o = (float4*)(Mc + (size_t)l * 16);   // 64B record
        o[0] = make_float4(sx * (1.0f / 64.0f), sy * (1.0f / 64.0f), sz * (1.0f / 64.0f), m00);
        o[1] = make_float4(m01, m02, m10, m11);
        o[2] = make_float4(m12, m20, m21, m22);
    }
}

// ---------------------------------------------------------------------------
// Kernel B: one thread per l.  3x3 SVD via Jacobi on M^T M, Kabsch sign fix,
// emit homogeneous 4x3 transform T:  out = [p,1] * T,  T = [R ; -c*R].
// ---------------------------------------------------------------------------
__global__ __launch_bounds__(256) void k_svd(const float* __restrict__ Mc,
                                             float* __restrict__ T) {
    const int l = blockIdx.x * 256 + threadIdx.x;
    const float4* q = (const float4*)(Mc + (size_t)l * 16);
    const float4 q0 = q[0], q1 = q[1], q2 = q[2];
    const float cx = q0.x, cy = q0.y, cz = q0.z;
    const float a00 = q0.w, a01 = q1.x, a02 = q1.y;
    const float a10 = q1.z, a11 = q1.w, a12 = q2.x;
    const float a20 = q2.y, a21 = q2.z, a22 = q2.w;

    // S = A^T A (symmetric)
    float s00 = a00 * a00 + a10 * a10 + a20 * a20;
    float s11 = a01 * a01 + a11 * a11 + a21 * a21;
    float s22 = a02 * a02 + a12 * a12 + a22 * a22;
    float s01 = a00 * a01 + a10 * a11 + a20 * a21;
    float s02 = a00 * a02 + a10 * a12 + a20 * a22;
    float s12 = a01 * a02 + a11 * a12 + a21 * a22;

    float v00 = 1.f, v01 = 0.f, v02 = 0.f;
    float v10 = 0.f, v11 = 1.f, v12 = 0.f;
    float v20 = 0.f, v21 = 0.f, v22 = 1.f;

#define ROT_(app, aqq, apq, arp, arq, p0, q0_, p1, q1_, p2, q2_)              \
    {                                                                         \
        const float apq_ = apq;                                               \
        if (__builtin_fabsf(apq_) > 1e-30f) {                                 \
            const float tau = (aqq - app) * (0.5f / apq_);                    \
            float tt = 1.0f / (__builtin_fabsf(tau) + __builtin_sqrtf(1.0f + tau * tau)); \
            if (tau < 0.0f) tt = -tt;                                         \
            const float cc = 1.0f / __builtin_sqrtf(1.0f + tt * tt);          \
            const float ss = tt * cc;                                         \
            app -= tt * apq_;  aqq += tt * apq_;  apq = 0.0f;                 \
            float t0_ = arp, t1_ = arq;                                       \
            arp = cc * t0_ - ss * t1_;  arq = ss * t0_ + cc * t1_;            \
            t0_ = p0;  t1_ = q0_; p0 = cc * t0_ - ss * t1_; q0_ = ss * t0_ + cc * t1_; \
            t0_ = p1;  t1_ = q1_; p1 = cc * t0_ - ss * t1_; q1_ = ss * t0_ + cc * t1_; \
            t0_ = p2;  t1_ = q2_; p2 = cc * t0_ - ss * t1_; q2_ = ss * t0_ + cc * t1_; \
        }                                                                     \
    }
#pragma unroll
    for (int sweep = 0; sweep < 4; ++sweep) {
        ROT_(s00, s11, s01, s02, s12, v00, v01, v10, v11, v20, v21);  // (0,1), r=2
        ROT_(s00, s22, s02, s01, s12, v00, v02, v10, v12, v20, v22);  // (0,2), r=1
        ROT_(s11, s22, s12, s01, s02, v01, v02, v11, v12, v21, v22);  // (1,2), r=0
    }
#undef ROT_

    // sort eigenpairs descending (columns of V)
    float l0 = s00, l1 = s11, l2 = s22;
#define CSWAP_(la, lb, pa0, pb0, pa1, pb1, pa2, pb2)                          \
    if (la < lb) {                                                            \
        float t_;                                                             \
        t_ = la; la = lb; lb = t_;                                            \
        t_ = pa0; pa0 = pb0; pb0 = t_;                                        \
        t_ = pa1; pa1 = pb1; pb1 = t_;                                        \
        t_ = pa2; pa2 = pb2; pb2 = t_;                                        \
    }
    CSWAP_(l0, l1, v00, v01, v10, v11, v20, v21);
    CSWAP_(l0, l2, v00, v02, v10, v12, v20, v22);
    CSWAP_(l1, l2, v01, v02, v11, v12, v21, v22);
#undef CSWAP_

    const float sg0 = __builtin_sqrtf(fmaxf(l0, 0.f));
    const float sg1 = __builtin_sqrtf(fmaxf(l1, 0.f));
    const float sg2 = __builtin_sqrtf(fmaxf(l2, 0.f));
    const float i0 = sg0 > 1e-12f ? 1.0f / sg0 : 0.0f;
    const float i1 = sg1 > 1e-12f ? 1.0f / sg1 : 0.0f;
    const float i2 = sg2 > 1e-12f ? 1.0f / sg2 : 0.0f;

    // u_i = A * v_i / sigma_i   (u{row}{i})
    const float u00 = (a00 * v00 + a01 * v10 + a02 * v20) * i0;
    const float u10 = (a10 * v00 + a11 * v10 + a12 * v20) * i0;
    const float u20 = (a20 * v00 + a21 * v10 + a22 * v20) * i0;
    const float u01 = (a00 * v01 + a01 * v11 + a02 * v21) * i1;
    const float u11 = (a10 * v01 + a11 * v11 + a12 * v21) * i1;
    const float u21 = (a20 * v01 + a21 * v11 + a22 * v21) * i1;
    const float u02 = (a00 * v02 + a01 * v12 + a02 * v22) * i2;
    const float u12 = (a10 * v02 + a11 * v12 + a12 * v22) * i2;
    const float u22 = (a20 * v02 + a21 * v12 + a22 * v22) * i2;

    const float det = a00 * (a11 * a22 - a12 * a21)
                    - a01 * (a10 * a22 - a12 * a20)
                    + a02 * (a10 * a21 - a11 * a20);
    const float sgn = det < 0.0f ? -1.0f : 1.0f;
    const float w02 = sgn * u02, w12 = sgn * u12, w22 = sgn * u22;

    // R[a][b] = sum_i u_i[a] * d_i * v_i[b],  v_i[b] = v{b}{i}
    const float r00 = u00 * v00 + u01 * v01 + w02 * v02;
    const float r01 = u00 * v10 + u01 * v11 + w02 * v12;
    const float r02 = u00 * v20 + u01 * v21 + w02 * v22;
    const float r10 = u10 * v00 + u11 * v01 + w12 * v02;
    const float r11 = u10 * v10 + u11 * v11 + w12 * v12;
    const float r12 = u10 * v20 + u11 * v21 + w12 * v22;
    const float r20 = u20 * v00 + u21 * v01 + w22 * v02;
    const float r21 = u20 * v10 + u21 * v11 + w22 * v12;
    const float r22 = u20 * v20 + u21 * v21 + w22 * v22;

    const float t0 = -(cx * r00 + cy * r10 + cz * r20);
    const float t1 = -(cx * r01 + cy * r11 + cz * r21);
    const float t2 = -(cx * r02 + cy * r12 + cz * r22);

    float4* o = (float4*)(T + (size_t)l * 16);        // 64B record, same flat order
    o[0] = make_float4(r00, r01, r02, r10);
    o[1] = make_float4(r11, r12, r20, r21);
    o[2] = make_float4(r22, t0, t1, t2);
}

// ---------------------------------------------------------------------------
// Kernel C: apply.  One wave per l, 16 chunks of 16 points, each chunk one
// v_wmma_f32_16x16x4_f32:  D(16x16) = A(16x4: T^T rows = out coord) x
// B(4x16: columns = homogeneous points).  Lane n<16 then holds point n's
// (x,y,z) in d[0..2] -> one contiguous 12B nontemporal b96 store per lane.
//
// A layout (32-bit 16x4): lanes 0-15 hold M=lane, VGPR0=K0, VGPR1=K1;
//                         lanes 16-31 hold M=lane-16, VGPR0=K2, VGPR1=K3.
// B layout (32-bit 4x16): lanes 0-15 hold N=lane, VGPR0=K0, VGPR1=K1;
//                         lanes 16-31 hold N=lane-16, VGPR0=K2, VGPR1=K3.
//
// Loads are branchless (cndmask offset / cndmask-with-1.0) and one chunk
// ahead (double buffer) so the next loads issue under the current WMMA.
// ---------------------------------------------------------------------------
__global__ __launch_bounds__(256) void k_apply(const float* __restrict__ x,
                                               const float* __restrict__ T,
                                               float* __restrict__ out) {
    const int lane = threadIdx.x & 31;
    const int l    = blockIdx.x * 8 + (threadIdx.x >> 5);
    const int m    = lane & 15;
    const bool lo  = lane < 16;
    const int koff = lo ? 0 : 2;          // K0 (=x) for lo lanes, K2 (=z) for hi

    // A = T^T padded to 16x4: A[m][k] = T[k][m] for m<3 (out coord), else 0.
    // Branchless: clamp row index, then zero with cndmask (T records are 64B,
    // so the clamped reads stay in-bounds).
    const float* t = T + (size_t)l * 16;
    const int mm = (m < 3) ? m : 0;
    float a0 = lo ? t[0 + mm] : t[6 + mm];
    float a1 = lo ? t[3 + mm] : t[9 + mm];
    if (m >= 3) { a0 = 0.0f; a1 = 0.0f; }
    const v2f A = {a0, a1};

    const float* xl = x + (size_t)l * (N_INP * 3);
    float*       ol = out + (size_t)l * (N_INP * 3);

    // prime chunk 0
    const float* p0 = xl + m * 3;
    float b0  = __builtin_nontemporal_load(p0 + koff);
    float py  = __builtin_nontemporal_load(p0 + 1);
    v2f B = {b0, lo ? py : 1.0f};

#pragma unroll
    for (int k = 0; k < 16; ++k) {
        v2f Bn = B;
        if (k < 15) {                     // resolved at compile time (unrolled)
            const float* pn = xl + ((k + 1) * 16 + m) * 3;
            const float nb0 = __builtin_nontemporal_load(pn + koff);
            const float npy = __builtin_nontemporal_load(pn + 1);
            Bn.x = nb0;
            Bn.y = lo ? npy : 1.0f;
        }
        v8f c = {};
        // 8 args: (neg_a, A, neg_b, B, c_mod, C, reuse_a, reuse_b)
        const v8f d = __builtin_amdgcn_wmma_f32_16x16x4_f32(
            false, A, false, B, (short)0, c, false, false);
        if (lo) {
            float* o = ol + (k * 16 + m) * 3;
            v3f val = {d[0], d[1], d[2]};          // M=0,1,2 -> out x,y,z
            __builtin_nontemporal_store(val, (v3f*)o);
        }
        B = Bn;
    }
}

// ---------------------------------------------------------------------------
// launch:  d_in[0]=x (L*256*3 f32), d_in[1]=ref_x (64*3 f32),
//          d_in[2]=align_idx (64 int), d_out = aligned_x (L*256*3 f32).
// ws layout: [0,768) refc | @1024 Mc (L*16 f32, 64B records)
//            | @1024+L*64 T (L*16 f32, 64B records).  ~8.4 MB total.
// ---------------------------------------------------------------------------
extern "C" void kernel_launch(void* const* d_in, const int* in_sizes, int n_in,
                              void* d_out, int out_size, void* d_ws, size_t ws_size,
                              hipStream_t stream) {
    const float* x     = (const float*)d_in[0];
    const float* ref_x = (const float*)d_in[1];
    const int*   aidx  = (const int*)d_in[2];
    float*       out   = (float*)d_out;

    char*  ws   = (char*)d_ws;
    float* refc = (float*)ws;
    float* Mc   = (float*)(ws + 1024);
    float* T    = (float*)(ws + 1024 + (size_t)L_TOT * 64);

    k_refc <<<1, 32, 0, stream>>>(ref_x, refc);
    k_cov  <<<L_TOT / 8, 256, 0, stream>>>(x, refc, aidx, Mc);
    k_svd  <<<L_TOT / 256, 256, 0, stream>>>(Mc, T);
    k_apply<<<L_TOT / 8, 256, 0, stream>>>(x, T, out);
}